// GraphClassifier_49392123904592
// MI455X (gfx1250) — compile-verified
//
#include <hip/hip_runtime.h>
#include <hip/hip_bf16.h>

#define N_NODES   100000
#define N_EDGES   3200000
#define N_GRAPHS  1024
#define F_IN      128
#define F_HID     64

typedef __attribute__((ext_vector_type(2))) float v2f;
typedef __attribute__((ext_vector_type(8))) float v8f;

// Native fire-and-forget fp32 atomic add (VGLOBAL atomic without return ->
// STOREcnt path, no CAS loop). Inline asm guarantees the hardware atomic.
__device__ __forceinline__ void atomAddF(float* p, float v) {
    asm volatile("global_atomic_add_f32 %0, %1, off"
                 :: "v"(p), "v"(v)
                 : "memory");
}

// ---------------------------------------------------------------------------
// K0: init deg=1 (self loop), zero pooled sums and counts
// ---------------------------------------------------------------------------
__global__ __launch_bounds__(256) void k_init(float* __restrict__ deg,
                                              float* __restrict__ sums,
                                              float* __restrict__ cnt) {
    int i = blockIdx.x * 256 + threadIdx.x;
    if (i < N_NODES)        deg[i]  = 1.0f;
    if (i < N_GRAPHS * 64)  sums[i] = 0.0f;
    if (i < N_GRAPHS)       cnt[i]  = 0.0f;
}

// ---------------------------------------------------------------------------
// K1: degree accumulation over edge destinations
// ---------------------------------------------------------------------------
__global__ __launch_bounds__(256) void k_edge_deg(const int* __restrict__ ei,
                                                  float* __restrict__ deg) {
    int e = blockIdx.x * 256 + threadIdx.x;
    if (e < N_EDGES) atomAddF(&deg[ei[N_EDGES + e]], 1.0f);
}

// ---------------------------------------------------------------------------
// K2: dinv = rsqrt(max(deg,1))
// ---------------------------------------------------------------------------
__global__ __launch_bounds__(256) void k_dinv(const float* __restrict__ deg,
                                              float* __restrict__ dinv) {
    int i = blockIdx.x * 256 + threadIdx.x;
    if (i < N_NODES) dinv[i] = rsqrtf(fmaxf(deg[i], 1.0f));
}

// ---------------------------------------------------------------------------
// K3: h = x @ W_gcn  via V_WMMA_F32_16X16X4_F32  (fp32, exact)
//
// One wave owns a 16-row x 64-col output strip: 4 accumulators, A fragment
// loaded once per k-step and reused across the 4 column-tile WMMAs. x is
// read exactly once (51 MB). W (32 KB) staged in LDS per block.
//
// A 16x4 f32 layout (ISA 7.12.2): lane l -> row (l&15), K pair k0+(l>>4)*2.
// B 4x16 mirrors with N=(l&15). C/D 16x16: lane l -> col (l&15),
// VGPR v -> row v + (l>>4)*8.
// ---------------------------------------------------------------------------
__global__ __launch_bounds__(256) void k_gemm(const float* __restrict__ x,
                                              const float* __restrict__ W,
                                              float* __restrict__ h) {
    __shared__ float sW[F_IN * F_HID];          // 32 KB
    for (int i = threadIdx.x; i < F_IN * F_HID; i += 256) sW[i] = W[i];
    __syncthreads();

    const int wave    = threadIdx.x >> 5;                 // 0..7: row strip in block
    const int lane    = threadIdx.x & 31;
    const int rowBase = (blockIdx.x * 8 + wave) * 16;
    if (rowBase >= N_NODES) return;                       // whole-wave exit only

    const int r     = lane & 15;
    const int khalf = (lane >> 4) * 2;                    // 0 or 2

    const float* xrow = x + (size_t)(rowBase + r) * F_IN;

    v8f acc0 = {}, acc1 = {}, acc2 = {}, acc3 = {};
    #pragma unroll 2
    for (int k0 = 0; k0 < F_IN; k0 += 4) {
        const int kb = k0 + khalf;
        v2f a = *(const v2f*)(xrow + kb);                 // contiguous, 8B aligned
        const float* w0 = &sW[(kb    ) * F_HID + r];
        const float* w1 = &sW[(kb + 1) * F_HID + r];
        v2f b0 = {w0[0],  w1[0]};
        v2f b1 = {w0[16], w1[16]};
        v2f b2 = {w0[32], w1[32]};
        v2f b3 = {w0[48], w1[48]};
        acc0 = __builtin_amdgcn_wmma_f32_16x16x4_f32(false, a, false, b0, (short)0, acc0, false, false);
        acc1 = __builtin_amdgcn_wmma_f32_16x16x4_f32(false, a, false, b1, (short)0, acc1, false, false);
        acc2 = __builtin_amdgcn_wmma_f32_16x16x4_f32(false, a, false, b2, (short)0, acc2, false, false);
        acc3 = __builtin_amdgcn_wmma_f32_16x16x4_f32(false, a, false, b3, (short)0, acc3, false, false);
    }

    const int rsel = (lane >> 4) * 8;
    float* hp = h + (size_t)rowBase * F_HID + r;
    #pragma unroll
    for (int v = 0; v < 8; ++v) {
        float* row = hp + (size_t)(v + rsel) * F_HID;
        row[0]  = acc0[v];
        row[16] = acc1[v];
        row[32] = acc2[v];
        row[48] = acc3[v];
    }
}

// ---------------------------------------------------------------------------
// K4: agg[i,:] = h[i,:] * dinv[i]^2   (self-loop term; also initializes agg)
// ---------------------------------------------------------------------------
__global__ __launch_bounds__(256) void k_selfloop(const float* __restrict__ h,
                                                  const float* __restrict__ dinv,
                                                  float* __restrict__ agg) {
    int idx = blockIdx.x * 256 + threadIdx.x;   // over N_NODES*64 = 6.4M
    if (idx < N_NODES * 64) {
        float d = dinv[idx >> 6];
        agg[idx] = h[idx] * d * d;
    }
}

// ---------------------------------------------------------------------------
// K5: edge scatter. One wave per edge; lane k moves float2 (features 2k,2k+1).
// agg[dst,:] += h[src,:] * dinv[src]*dinv[dst]   (1.7 GB total, L2-resident)
// ---------------------------------------------------------------------------
__global__ __launch_bounds__(256) void k_scatter(const int* __restrict__ ei,
                                                 const float* __restrict__ dinv,
                                                 const float* __restrict__ h,
                                                 float* __restrict__ agg) {
    const int e = (blockIdx.x * 256 + threadIdx.x) >> 5;
    if (e >= N_EDGES) return;
    const int lane = threadIdx.x & 31;
    const int src  = ei[e];
    const int dst  = ei[N_EDGES + e];
    const float nrm = dinv[src] * dinv[dst];
    v2f v = ((const v2f*)(h + (size_t)src * F_HID))[lane];
    float* ap = agg + (size_t)dst * F_HID + lane * 2;
    atomAddF(ap,     v.x * nrm);
    atomAddF(ap + 1, v.y * nrm);
}

// ---------------------------------------------------------------------------
// K6: per-node: relu(agg + b_gcn) accumulated into per-graph sums + counts.
// One wave per node, lane moves float2.
// ---------------------------------------------------------------------------
__global__ __launch_bounds__(256) void k_pool(const float* __restrict__ agg,
                                              const float* __restrict__ bgcn,
                                              const int* __restrict__ batch,
                                              float* __restrict__ sums,
                                              float* __restrict__ cnt) {
    const int i = (blockIdx.x * 256 + threadIdx.x) >> 5;
    if (i >= N_NODES) return;
    const int lane = threadIdx.x & 31;
    const int g = batch[i];
    v2f v = ((const v2f*)(agg + (size_t)i * F_HID))[lane];
    float r0 = fmaxf(v.x + bgcn[lane * 2],     0.0f);
    float r1 = fmaxf(v.y + bgcn[lane * 2 + 1], 0.0f);
    float* sp = sums + (size_t)g * F_HID + lane * 2;
    atomAddF(sp,     r0);
    atomAddF(sp + 1, r1);
    if (lane == 0) atomAddF(&cnt[g], 1.0f);
}

// ---------------------------------------------------------------------------
// K7: per-graph MLP head: relu(pooled@W1+b1)@W2+b2. One 64-thread block/graph.
// ---------------------------------------------------------------------------
__global__ __launch_bounds__(64) void k_mlp(const float* __restrict__ sums,
                                            const float* __restrict__ cnt,
                                            const float* __restrict__ W1,
                                            const float* __restrict__ b1,
                                            const float* __restrict__ W2,
                                            const float* __restrict__ b2,
                                            float* __restrict__ out) {
    __shared__ float p[F_HID];
    __shared__ float hid[32];
    const int g = blockIdx.x, t = threadIdx.x;
    const float inv = 1.0f / fmaxf(cnt[g], 1.0f);
    p[t] = sums[(size_t)g * F_HID + t] * inv;
    __syncthreads();
    if (t < 32) {
        float a = b1[t];
        #pragma unroll 8
        for (int f = 0; f < F_HID; ++f) a += p[f] * W1[f * 32 + t];
        hid[t] = fmaxf(a, 0.0f);
    }
    __syncthreads();
    if (t < 4) {
        float a = b2[t];
        #pragma unroll
        for (int j = 0; j < 32; ++j) a += hid[j] * W2[j * 4 + t];
        out[(size_t)g * 4 + t] = a;
    }
}

// ---------------------------------------------------------------------------
extern "C" void kernel_launch(void* const* d_in, const int* in_sizes, int n_in,
                              void* d_out, int out_size, void* d_ws, size_t ws_size,
                              hipStream_t stream) {
    const float* x     = (const float*)d_in[0];
    const int*   ei    = (const int*)  d_in[1];
    const int*   batch = (const int*)  d_in[2];
    const float* Wg    = (const float*)d_in[3];
    const float* bg    = (const float*)d_in[4];
    const float* W1    = (const float*)d_in[5];
    const float* b1    = (const float*)d_in[6];
    const float* W2    = (const float*)d_in[7];
    const float* b2    = (const float*)d_in[8];
    float* out = (float*)d_out;

    // Workspace carve-up (all fp32). Total ~52 MB.
    float* ws   = (float*)d_ws;
    float* deg  = ws;                                  // [N]
    float* dinv = deg  + N_NODES;                      // [N]
    float* h    = dinv + N_NODES;                      // [N,64]
    float* agg  = h    + (size_t)N_NODES * F_HID;      // [N,64]
    float* sums = agg  + (size_t)N_NODES * F_HID;      // [G,64]
    float* cnt  = sums + (size_t)N_GRAPHS * F_HID;     // [G]

    const int nb_nodes = (N_NODES + 255) / 256;
    const int nb_edges = (N_EDGES + 255) / 256;
    const int nb_feat  = (N_NODES * 64 + 255) / 256;
    const int nb_escat = (int)(((size_t)N_EDGES * 32 + 255) / 256);
    const int nb_npool = (int)(((size_t)N_NODES * 32 + 255) / 256);
    const int nb_gemm  = (N_NODES / 16 + 7) / 8;       // 8 wave-strips per block

    k_init    <<<nb_nodes, 256, 0, stream>>>(deg, sums, cnt);
    k_edge_deg<<<nb_edges, 256, 0, stream>>>(ei, deg);
    k_dinv    <<<nb_nodes, 256, 0, stream>>>(deg, dinv);
    k_gemm    <<<nb_gemm,  256, 0, stream>>>(x, Wg, h);
    k_selfloop<<<nb_feat,  256, 0, stream>>>(h, dinv, agg);
    k_scatter <<<nb_escat, 256, 0, stream>>>(ei, dinv, h, agg);
    k_pool    <<<nb_npool, 256, 0, stream>>>(agg, bg, batch, sums, cnt);
    k_mlp     <<<N_GRAPHS, 64, 0, stream>>>(sums, cnt, W1, b1, W2, b2, out);
}